// PointLoss_1709396983865
// MI455X (gfx1250) — compile-verified
//
#include <hip/hip_runtime.h>
#include <hip/hip_bf16.h>
#include <math.h>

// CDNA5 / gfx1250. wave32. One wave per (b, hw) cell; 8 cells per 256-thread block.
// B=4, H=W=64, M=8, K=64, T=64, cells = 16384, blocks = 2048.

typedef __attribute__((ext_vector_type(2))) float v2f;
typedef __attribute__((ext_vector_type(8))) float v8f;

#define NCELLS 16384
#define NBLOCKS 2048
#define BIGCOST 1e9f

__device__ __forceinline__ float log1pexp(float x) {
    // log(1 + e^x), stable
    return (x > 0.f) ? (x + log1pf(expf(-x))) : log1pf(expf(x));
}

__global__ __launch_bounds__(256) void pl_cells_kernel(
    const float* __restrict__ ps,   // [B,HW,K]   = [cell,64]
    const float* __restrict__ po,   // [B,HW,K,2] = [cell,64,2]
    const int*   __restrict__ tg,   // [B,512,512]
    float* __restrict__ blkF,       // [NBLOCKS] obj + 10*loc partial
    int*   __restrict__ blkI)       // [NBLOCKS] num_pos partial
{
    __shared__ int   sTile[8 * 64];   // 8 target rows x 64 cols (2048 B)
    __shared__ float sRedF[8];
    __shared__ int   sRedI[8];

    const int tid  = threadIdx.x;
    const int lane = tid & 31;
    const int wv   = tid >> 5;

    const int cell0 = blockIdx.x * 8;       // 8 consecutive cells share b,h and w0..w0+7
    const int b  = cell0 >> 12;             // cell0 / 4096
    const int n0 = cell0 & 4095;
    const int h  = n0 >> 6;
    const int w0 = n0 & 63;                 // multiple of 8

    // ---- gfx1250 async global->LDS: one contiguous 8x256B region for all 8 cells ----
    // rows h*8 .. h*8+7 ; cols w0*8 .. w0*8+63 of the 512x512 target plane
    const int* gbase = tg + (((size_t)b * 512 + (size_t)h * 8) * 512 + (size_t)w0 * 8);
    if (tid < 32) {
        // 2048 B = 128 x b128 chunks; 4 chunks per lane
        #pragma unroll
        for (int c = lane; c < 128; c += 32) {
            const int row = c >> 4, col16 = c & 15;
            unsigned long long ga = (unsigned long long)(gbase + row * 512 + col16 * 4);
            // flat LDS pointer's low 32 bits == LDS byte offset
            unsigned ldsOff = (unsigned)(unsigned long long)(&sTile[c * 4]);
            asm volatile("global_load_async_to_lds_b128 %0, %1, off"
                         :: "v"(ldsOff), "v"(ga) : "memory");
        }
    }
    asm volatile("s_wait_asynccnt 0" ::: "memory");
    __syncthreads();

    const int cell = cell0 + wv;

    // ---- per-cell target bits: t = m1*8 + m2, lane owns t=lane and t=lane+32 ----
    const int m2a = lane & 7,        m1a = lane >> 3;        // t = lane   -> m1 in 0..3
    const int m2b = m2a,             m1b = (lane + 32) >> 3; // t = lane+32 -> m1 in 4..7
    const int tv0 = sTile[m1a * 64 + wv * 8 + m2a];
    const int tv1 = sTile[m1b * 64 + wv * 8 + m2b];
    const unsigned mlo = __builtin_amdgcn_ballot_w32(tv0 > 0);
    const unsigned mhi = __builtin_amdgcn_ballot_w32(tv1 > 0);
    const int npos = __popc(mlo) + __popc(mhi);

    // ---- per-lane prediction data for k = lane, k = lane+32 ----
    const int kbase = cell * 64;
    const float x0  = ps[kbase + lane];
    const float x1  = ps[kbase + lane + 32];
    const float py0 = po[(kbase + lane) * 2 + 0];
    const float px0 = po[(kbase + lane) * 2 + 1];
    const float py1 = po[(kbase + lane + 32) * 2 + 0];
    const float px1 = po[(kbase + lane + 32) * 2 + 1];

    const float s0 = 1.f / (1.f + expf(-x0));
    const float s1 = 1.f / (1.f + expf(-x1));
    // cls cost for positives is (1 - sigma)^0.2, constant over t -> hoist
    const float clsp0 = powf(fmaxf(1.f - s0, 0.f), 0.2f);
    const float clsp1 = powf(fmaxf(1.f - s1, 0.f), 0.2f);

    // ---- greedy assignment over positives (ascending t == stable-desc sort order) ----
    bool  a0 = false, a1 = false;
    float tg0 = 0.f, tg1 = 0.f;
    float locAcc = 0.f;

    unsigned long long mm = ((unsigned long long)mhi << 32) | (unsigned long long)mlo;
    while (mm) {
        const int t = __ffsll(mm) - 1;
        mm &= (mm - 1);
        const float offy = ((float)(t >> 3) - 3.5f) * 0.125f;
        const float offx = ((float)(t & 7)  - 3.5f) * 0.125f;

        const float l0 = fabsf(py0 - offy) + fabsf(px0 - offx);
        const float l1 = fabsf(py1 - offy) + fabsf(px1 - offx);
        const float c0 = a0 ? BIGCOST : powf(l0, 0.8f) * clsp0;
        const float c1 = a1 ? BIGCOST : powf(l1, 0.8f) * clsp1;

        // per-lane best, tie -> lower k (slot0)
        float cb; int kb;
        if (c1 < c0) { cb = c1; kb = lane + 32; } else { cb = c0; kb = lane; }
        // wave32 butterfly argmin, tie -> lowest k (matches jnp.argmin)
        #pragma unroll
        for (int off = 16; off > 0; off >>= 1) {
            const float oc = __shfl_xor(cb, off, 32);
            const int   ok = __shfl_xor(kb, off, 32);
            if (oc < cb || (oc == cb && ok < kb)) { cb = oc; kb = ok; }
        }
        if (kb == lane)           { a0 = true; tg0 = 1.f; locAcc += l0; }
        else if (kb == lane + 32) { a1 = true; tg1 = 1.f; locAcc += l1; }
    }

    // ---- focal loss for this lane's two k slots ----
    float objAcc = 0.f;
    {
        const float L0 = log1pexp(x0);                 // ce = L - t*x
        const float ce0 = L0 - tg0 * x0;
        const float pt0 = (tg0 > 0.f) ? s0 : (1.f - s0);
        objAcc += ce0 * (1.f - pt0) * (1.f - pt0) * ((tg0 > 0.f) ? 0.6f : 0.4f);
        const float L1 = log1pexp(x1);
        const float ce1 = L1 - tg1 * x1;
        const float pt1 = (tg1 > 0.f) ? s1 : (1.f - s1);
        objAcc += ce1 * (1.f - pt1) * (1.f - pt1) * ((tg1 > 0.f) ? 0.6f : 0.4f);
    }
    const float part = objAcc + 10.f * locAcc;

    // ---- wave-wide f32 reduction through the matrix unit ----
    // A (16x4 f32): each lane's 2 slots hold (part, 0), one partial per lane; B = all-ones.
    // D rows replicate row-sums; lane sums its 8 D regs (rows m=0..7 or 8..15),
    // one shfl_xor(16) gives the cell total on every lane.
    v2f a; a.x = part; a.y = 0.f;
    v2f bb; bb.x = 1.f; bb.y = 1.f;
    v8f c = {};
    v8f d = __builtin_amdgcn_wmma_f32_16x16x4_f32(
        /*neg_a=*/false, a, /*neg_b=*/false, bb,
        /*c_mod=*/(short)0, c, /*reuse_a=*/false, /*reuse_b=*/false);
    float rsum = d[0] + d[1] + d[2] + d[3] + d[4] + d[5] + d[6] + d[7];
    const float waveTot = rsum + __shfl_xor(rsum, 16, 32);

    if (lane == 0) { sRedF[wv] = waveTot; sRedI[wv] = npos; }
    __syncthreads();
    if (tid == 0) {
        float f = 0.f; int ii = 0;
        #pragma unroll
        for (int i = 0; i < 8; ++i) { f += sRedF[i]; ii += sRedI[i]; }
        blkF[blockIdx.x] = f;
        blkI[blockIdx.x] = ii;
    }
}

__global__ __launch_bounds__(256) void pl_reduce_kernel(
    const float* __restrict__ blkF, const int* __restrict__ blkI,
    float* __restrict__ out, int nblk)
{
    __shared__ float sf[256];
    __shared__ int   si[256];
    const int tid = threadIdx.x;
    float f = 0.f; int c = 0;
    for (int i = tid; i < nblk; i += 256) { f += blkF[i]; c += blkI[i]; }
    sf[tid] = f; si[tid] = c;
    __syncthreads();
    for (int s = 128; s > 0; s >>= 1) {
        if (tid < s) { sf[tid] += sf[tid + s]; si[tid] += si[tid + s]; }
        __syncthreads();
    }
    if (tid == 0) out[0] = sf[0] / fmaxf((float)si[0], 1.0f);
}

extern "C" void kernel_launch(void* const* d_in, const int* in_sizes, int n_in,
                              void* d_out, int out_size, void* d_ws, size_t ws_size,
                              hipStream_t stream) {
    const float* ps = (const float*)d_in[0];   // [4,4096,64]
    const float* po = (const float*)d_in[1];   // [4,4096,64,2]
    const int*   tg = (const int*)d_in[2];     // [4,512,512]
    (void)in_sizes; (void)n_in; (void)out_size; (void)ws_size;

    float* blkF = (float*)d_ws;                // NBLOCKS floats
    int*   blkI = (int*)((char*)d_ws + NBLOCKS * sizeof(float));

    pl_cells_kernel<<<NBLOCKS, 256, 0, stream>>>(ps, po, tg, blkF, blkI);
    pl_reduce_kernel<<<1, 256, 0, stream>>>(blkF, blkI, (float*)d_out, NBLOCKS);
}